// TokenInterNeck2_10548439679315
// MI455X (gfx1250) — compile-verified
//
#include <hip/hip_runtime.h>
#include <cstdint>
#include <cstddef>

// Problem dims (from reference setup_inputs; SCALE_FACTOR=2 <= scale_orig=4
// so H = W = 128*2 = 256). All powers of two except the cell stride.
#define B_   8
#define N_   1024     // tokens (2^10)
#define C_   128      // channels (2^7)
#define N0_  16384    // original positions per batch (2^14)
#define H_   256
#define W_   256
#define CS_  132      // accumulator cell stride in floats: 128 sums + count + pad (528B, 16B-aligned)
#define EPSF 1e-6f

// ---------------------------------------------------------------------------
// Cell index: clip loc to [-1,1], normalized -> pixel with round-half-even
// (v_rndne_f32 via rintf, matches jnp.round / torch.round), clamp to grid.
// ---------------------------------------------------------------------------
__device__ __forceinline__ int cell_of(float lx, float ly) {
    lx = fminf(fmaxf(lx, -1.0f), 1.0f);
    ly = fminf(fmaxf(ly, -1.0f), 1.0f);
    float fx = rintf(0.5f * (lx + 1.0f) * (float)W_ - 0.5f);
    float fy = rintf(0.5f * (ly + 1.0f) * (float)H_ - 0.5f);
    int ix = min(max((int)fx, 0), W_ - 1);
    int iy = min(max((int)fy, 0), H_ - 1);
    return iy * W_ + ix;
}

// ---------------------------------------------------------------------------
// Zero-fill (float4 grid-stride). Regular-temporal stores: for the per-batch
// accumulator slice (34.6MB) these stay resident in the 192MB L2.
// ---------------------------------------------------------------------------
__global__ void zero_f4(float4* __restrict__ p, size_t n4) {
    size_t i = (size_t)blockIdx.x * blockDim.x + threadIdx.x;
    size_t stride = (size_t)gridDim.x * blockDim.x;
    float4 z = make_float4(0.f, 0.f, 0.f, 0.f);
    for (; i < n4; i += stride) p[i] = z;
}

// ---------------------------------------------------------------------------
// Per-batch scatter: one wave32 per point. Lane l handles channels 4l..4l+3.
// Slice layout [H,W,CS_]: per-point atomics hit 516 contiguous bytes ->
// cacheline-coalesced global_atomic_add_f32 (no-return via unsafeAtomicAdd),
// all RMW in L2 (slice is resident after the zero pass).
// ---------------------------------------------------------------------------
__global__ void scatter_agg(const float4* __restrict__ x4b,   // batch's [N,C]
                            const float2* __restrict__ loc2b, // batch's [N0,2]
                            const int*    __restrict__ idxb,  // batch's [N0]
                            float*        __restrict__ agg) { // [H*W, CS_]
    int tid  = threadIdx.x;
    int lane = tid & 31;
    int p    = blockIdx.x * (256 / 32) + (tid >> 5);   // point in [0, N0)
    float2 lo = loc2b[p];                              // broadcast load per wave
    int tok   = idxb[p];
    int cell  = cell_of(lo.x, lo.y);
    float* cbase = agg + (size_t)cell * CS_;
    float4 v = x4b[(size_t)tok * (C_ / 4) + lane];     // 512B coalesced per wave
    float* dst = cbase + lane * 4;
    unsafeAtomicAdd(dst + 0, v.x);
    unsafeAtomicAdd(dst + 1, v.y);
    unsafeAtomicAdd(dst + 2, v.z);
    unsafeAtomicAdd(dst + 3, v.w);
    if (lane == 0) unsafeAtomicAdd(cbase + 128, 1.0f);
}

// ---------------------------------------------------------------------------
// Per-batch finalize + transpose: block = (h, 32-wide w tile).
// Stage 32 cells x 132 floats (16.9KB) into LDS with gfx1250 async-to-LDS
// (global_load_async_to_lds_b128, ASYNCcnt) using th:TH_LOAD_LU — last-use
// discards the dirty accumulator lines in cache, so the slice is never
// written back to HBM. Output stores are non-temporal (TH=NT) so the 256MB
// stream doesn't evict the hot slice.
// ---------------------------------------------------------------------------
__global__ void finalize_t(const float* __restrict__ agg, float* __restrict__ outb) {
    __shared__ float tile[32 * CS_];
    int bid = blockIdx.x;                // [0, H_*(W_/32))
    int w0  = (bid & 7) * 32;            // W_/32 = 8 tiles
    int h   = bid >> 3;
    const float* gbase = agg + (size_t)((h << 8) + w0) * CS_;
    uint32_t lds_base = (uint32_t)(size_t)(&tile[0]);  // LDS byte offset (addr[31:0])
    // 32 cells * 33 float4 = 1056 16-byte async transfers
    for (int t = threadIdx.x; t < 32 * (CS_ / 4); t += blockDim.x) {
        uint64_t gaddr = (uint64_t)(size_t)(gbase + t * 4);
        uint32_t laddr = lds_base + (uint32_t)(t * 16);
        asm volatile("global_load_async_to_lds_b128 %0, %1, off th:TH_LOAD_LU"
                     :: "v"(laddr), "v"(gaddr) : "memory");
    }
    asm volatile("s_wait_asynccnt 0" ::: "memory");
    __syncthreads();
    int w  = threadIdx.x & 31;           // consecutive lanes -> consecutive w
    int c0 = threadIdx.x >> 5;           // 0..7
    float inv = 1.0f / (tile[w * CS_ + 128] + EPSF);
    float* obase = outb + ((size_t)h << 8) + (w0 + w); // + c*H*W per channel
    #pragma unroll
    for (int c = c0; c < C_; c += 8)
        __builtin_nontemporal_store(tile[w * CS_ + c] * inv,
                                    obase + ((size_t)c << 16)); // 128B/wave stores
}

// ---------------------------------------------------------------------------
// Fallback path (workspace smaller than one 34.6MB slice): accumulate
// directly in d_out ([B,C,H,W], strided atomics) + counts in d_ws (2MB).
// ---------------------------------------------------------------------------
__global__ void scatter_direct(const float4* __restrict__ x4,
                               const float2* __restrict__ loc2,
                               const int*    __restrict__ idx_agg,
                               float*        __restrict__ out,
                               float*        __restrict__ counts) {
    int tid  = threadIdx.x;
    int lane = tid & 31;
    int p    = blockIdx.x * (256 / 32) + (tid >> 5);   // point in [0, B*N0)
    int b    = p >> 14;                                // / N0_
    float2 lo = loc2[p];
    int tok   = idx_agg[p];
    int cell  = cell_of(lo.x, lo.y);
    float4 v = x4[(size_t)((b << 10) + tok) * (C_ / 4) + lane];
    float* obase = out + ((size_t)b << 23) + cell;     // + c*H*W per channel
    int c = lane * 4;
    unsafeAtomicAdd(obase + ((size_t)(c + 0) << 16), v.x);
    unsafeAtomicAdd(obase + ((size_t)(c + 1) << 16), v.y);
    unsafeAtomicAdd(obase + ((size_t)(c + 2) << 16), v.z);
    unsafeAtomicAdd(obase + ((size_t)(c + 3) << 16), v.w);
    if (lane == 0) unsafeAtomicAdd(counts + (b << 16) + cell, 1.0f);
}

__global__ void finalize_direct(float* __restrict__ out, const float* __restrict__ counts) {
    size_t n4 = ((size_t)B_ * C_ * H_ * W_) >> 2;      // 2^24 float4
    size_t stride = (size_t)gridDim.x * blockDim.x;
    const float4* cv4 = (const float4*)counts;
    float4* o4 = (float4*)out;
    for (size_t i4 = (size_t)blockIdx.x * blockDim.x + threadIdx.x; i4 < n4; i4 += stride) {
        size_t li = i4 << 2;                           // flat [B,C,H,W] index
        int b  = (int)(li >> 23);
        int hw = (int)(li & 0xFFFF);                   // h*W + w (w multiple of 4)
        float4 cv = cv4[((size_t)(b << 16) + hw) >> 2];
        float4 v = o4[i4];
        v.x *= 1.0f / (cv.x + EPSF);
        v.y *= 1.0f / (cv.y + EPSF);
        v.z *= 1.0f / (cv.z + EPSF);
        v.w *= 1.0f / (cv.w + EPSF);
        o4[i4] = v;
    }
}

// ---------------------------------------------------------------------------
// Launch: per-batch (zero -> scatter -> finalize) so the 34.6MB accumulator
// slice stays L2-resident for its whole lifetime. All on `stream`
// (graph-capture safe), not bitwise deterministic (float atomics).
// ---------------------------------------------------------------------------
extern "C" void kernel_launch(void* const* d_in, const int* in_sizes, int n_in,
                              void* d_out, int out_size, void* d_ws, size_t ws_size,
                              hipStream_t stream) {
    const float4* x4       = (const float4*)d_in[0];  // [B,N,C] f32
    // d_in[1] = loc (unused by this branch of the reference)
    const float2* loc2     = (const float2*)d_in[2];  // [B,N0,2] f32
    const int*    idx_agg  = (const int*)d_in[3];     // [B,N0] i32
    // d_in[4]/d_in[5] = map_h/map_w (128/128, dims baked in above)
    float* out = (float*)d_out;

    const size_t sliceBytes = (size_t)H_ * W_ * CS_ * 4;   // 34.6 MB per batch

    if (ws_size >= sliceBytes) {
        float* agg = (float*)d_ws;
        for (int b = 0; b < B_; ++b) {
            zero_f4<<<2048, 256, 0, stream>>>((float4*)agg, sliceBytes >> 4);
            scatter_agg<<<N0_ / 8, 256, 0, stream>>>(
                x4 + (size_t)b * N_ * (C_ / 4),
                loc2 + (size_t)b * N0_,
                idx_agg + (size_t)b * N0_,
                agg);
            finalize_t<<<H_ * (W_ / 32), 256, 0, stream>>>(agg, out + ((size_t)b << 23));
        }
    } else {
        float* counts = (float*)d_ws;                      // needs 2 MB
        zero_f4<<<2048, 256, 0, stream>>>((float4*)out, ((size_t)B_ * C_ * H_ * W_) >> 2);
        zero_f4<<<256, 256, 0, stream>>>((float4*)counts, ((size_t)B_ * H_ * W_) >> 2);
        scatter_direct<<<(B_ * N0_) / 8, 256, 0, stream>>>(x4, loc2, idx_agg, out, counts);
        finalize_direct<<<4096, 256, 0, stream>>>(out, counts);
    }
}